// ResGCN_2576980377707
// MI455X (gfx1250) — compile-verified
//
#include <hip/hip_runtime.h>
#include <hip/hip_bf16.h>

typedef __attribute__((ext_vector_type(2))) float v2f;
typedef __attribute__((ext_vector_type(8))) float v8f;

#define GN   50000
#define GE   640000
#define GIN  256
#define GH   128
#define GOUT 40
#define GEPS 1e-5f
#define KCHUNK 64

// ---------------------------------------------------------------- utilities
__global__ void k_fill(float* __restrict__ p, float v, int n) {
  int i = blockIdx.x * blockDim.x + threadIdx.x;
  if (i < n) p[i] = v;
}

// degree of (A + 2I): deg buffer pre-filled with 2.0, add 1 per incoming edge
__global__ void k_deg(const int* __restrict__ ei, float* __restrict__ deg) {
  int e = blockIdx.x * blockDim.x + threadIdx.x;
  if (e < GE) atomicAdd(&deg[ei[GE + e]], 1.0f);
}

__global__ void k_rsqrt_inplace(float* __restrict__ d) {
  int i = blockIdx.x * blockDim.x + threadIdx.x;
  if (i < GN) d[i] = rsqrtf(d[i]);
}

// ------------------------------------------------- fp32 WMMA GEMM: Y = X*W(+bias)
// X: [nrows, K] row-major, W: [K, ncols] row-major, Y: [nrows, ncols]
// NC = padded column count (multiple of 16, >= ncols). One wave -> 16 x NC strip.
template <int NC>
__global__ __launch_bounds__(256) void k_gemm_wmma(
    const float* __restrict__ X, const float* __restrict__ W,
    const float* __restrict__ bias, float* __restrict__ Y,
    int nrows, int K, int ncols) {
  // pair-interleaved weight panel: lds2[pair*NC + n] = {W[2p][n], W[2p+1][n]}
  __shared__ float lds[KCHUNK * NC];
  constexpr int NT = NC / 16;

  const int tid  = threadIdx.x;
  const int lane = tid & 31;
  const int wave = tid >> 5;
  const int l15  = lane & 15;
  const int lhi  = lane >> 4;  // 0: K={k,k+1}, 1: K={k+2,k+3}
  const int rowbase = blockIdx.x * 128 + wave * 16;

  int arow = rowbase + l15;
  if (arow >= nrows) arow = nrows - 1;  // clamp (store is guarded)
  const float* xrow = X + (size_t)arow * K + lhi * 2;

  v8f acc[NT] = {};
  const v2f* lds2 = (const v2f*)lds;

  for (int k0 = 0; k0 < K; k0 += KCHUNK) {
    __syncthreads();
    for (int idx = tid; idx < KCHUNK * NC; idx += 256) {
      int kk = idx / NC;
      int n  = idx - kk * NC;
      float v = (n < ncols) ? W[(size_t)(k0 + kk) * ncols + n] : 0.0f;
      lds[(((kk >> 1) * NC + n) << 1) | (kk & 1)] = v;
    }
    __syncthreads();

    // software-pipeline the A fragment one K-step ahead
    v2f a = *(const v2f*)(xrow + k0);
#pragma unroll
    for (int ks = 0; ks < KCHUNK; ks += 4) {
      v2f a_cur = a;
      if (ks + 4 < KCHUNK) a = *(const v2f*)(xrow + k0 + ks + 4);

      // gather all B fragments first so DS loads batch under one wait
      v2f bf[NT];
      int bbase = ((ks >> 1) + lhi) * NC + l15;
#pragma unroll
      for (int t = 0; t < NT; ++t) bf[t] = lds2[bbase + t * 16];

#pragma unroll
      for (int t = 0; t < NT; ++t)
        acc[t] = __builtin_amdgcn_wmma_f32_16x16x4_f32(
            false, a_cur, false, bf[t], (short)0, acc[t], false, false);
    }
  }

#pragma unroll
  for (int t = 0; t < NT; ++t) {
    int col = t * 16 + l15;
    float badd = (bias != nullptr && col < ncols) ? bias[col] : 0.0f;
#pragma unroll
    for (int v = 0; v < 8; ++v) {
      int row = rowbase + v + lhi * 8;  // C/D layout: VGPR v -> rows v / v+8
      if (row < nrows && col < ncols)
        Y[(size_t)row * ncols + col] = acc[t][v] + badd;
    }
  }
}

// ----------------------------------------- edge scatter: agg[c] += norm * h[r]
__global__ __launch_bounds__(256) void k_scatter(
    const float* __restrict__ h, const int* __restrict__ ei,
    const float* __restrict__ dis, float* __restrict__ agg) {
  int e = blockIdx.x * 8 + (threadIdx.x >> 5);
  if (e >= GE) return;
  int lane = threadIdx.x & 31;
  int r = ei[e];
  int c = ei[GE + e];
  float nrm = dis[r] * dis[c];
  const float* hr = h + (size_t)r * GH;
  float* ac = agg + (size_t)c * GH;
#pragma unroll
  for (int j = 0; j < GH; j += 32)
    atomicAdd(&ac[j + lane], nrm * hr[j + lane]);
}

// ---------- self-loop + bias finish, fused with BN statistics (sum, sumsq)
__global__ __launch_bounds__(128) void k_finish_stats(
    float* __restrict__ agg, const float* __restrict__ h,
    const float* __restrict__ dis, const float* __restrict__ bias,
    float* __restrict__ stats, int nrows) {
  int j = threadIdx.x;       // channel 0..127
  int r0 = blockIdx.x * 32;
  float b = bias[j];
  float s = 0.0f, s2 = 0.0f;
  for (int i = 0; i < 32; ++i) {
    int r = r0 + i;
    if (r >= nrows) break;   // uniform across block
    float d = dis[r];
    size_t o = (size_t)r * GH + j;
    float v = agg[o] + (2.0f * d * d) * h[o] + b;
    agg[o] = v;
    s += v;
    s2 += v * v;
  }
  atomicAdd(&stats[j], s);
  atomicAdd(&stats[GH + j], s2);
}

// ------------------------------- BN (batch stats) + ReLU + residual add
__global__ __launch_bounds__(256) void k_bn_relu_add(
    const float* __restrict__ conv, const float* __restrict__ stats,
    const float* __restrict__ g, const float* __restrict__ be,
    const float* res, float* out, int nrows) {
  int idx = blockIdx.x * blockDim.x + threadIdx.x;
  int total = nrows * GH;
  if (idx >= total) return;
  int j = idx & (GH - 1);
  float invN = 1.0f / (float)nrows;
  float m = stats[j] * invN;
  float var = stats[GH + j] * invN - m * m;
  float inv = rsqrtf(var + GEPS);
  float v = (conv[idx] - m) * inv * g[j] + be[j];
  v = v > 0.0f ? v : 0.0f;
  out[idx] = v + res[idx];
}

// ---------------------------------------------------------------- launcher
extern "C" void kernel_launch(void* const* d_in, const int* in_sizes, int n_in,
                              void* d_out, int out_size, void* d_ws, size_t ws_size,
                              hipStream_t stream) {
  const float* x   = (const float*)d_in[0];
  const int*   ei  = (const int*)d_in[1];
  const float* W0  = (const float*)d_in[2];
  const float* b0  = (const float*)d_in[3];
  const float* g0  = (const float*)d_in[4];
  const float* be0 = (const float*)d_in[5];
  const float* pW  = (const float*)d_in[6];
  const float* pb  = (const float*)d_in[7];
  const float* W1  = (const float*)d_in[8];
  const float* b1  = (const float*)d_in[9];
  const float* g1  = (const float*)d_in[10];
  const float* be1 = (const float*)d_in[11];
  const float* hW  = (const float*)d_in[12];
  const float* hb  = (const float*)d_in[13];
  float* out = (float*)d_out;

  const size_t NH = (size_t)GN * GH;              // 6.4M floats
  float* ws    = (float*)d_ws;
  float* bufA  = ws;                              // h0 / h1
  float* bufB  = ws + NH;                         // res_proj -> x1 -> x2
  float* bufC  = ws + 2 * NH;                     // agg0 / agg1
  float* dis   = ws + 3 * NH;                     // deg -> rsqrt(deg), N floats
  float* stats = ws + 3 * NH + GN;                // 2*GH floats

  const int gNH   = (int)((NH + 255) / 256);
  const int gN    = (GN + 255) / 256;
  const int gGemm = (GN + 127) / 128;
  const int gEdge = (GE + 7) / 8;
  const int gFin  = (GN + 31) / 32;

  // ---- degree / normalization (shared by both convs)
  k_fill<<<gN, 256, 0, stream>>>(dis, 2.0f, GN);
  k_deg<<<(GE + 255) / 256, 256, 0, stream>>>(ei, dis);
  k_rsqrt_inplace<<<gN, 256, 0, stream>>>(dis);

  // ---- block 0: conv(x,W0) -> bn -> relu -> + (x@pW + pb)
  k_fill<<<gNH, 256, 0, stream>>>(bufC, 0.0f, (int)NH);
  k_fill<<<1, 256, 0, stream>>>(stats, 0.0f, 2 * GH);
  k_gemm_wmma<128><<<gGemm, 256, 0, stream>>>(x, W0, nullptr, bufA, GN, GIN, GH);
  k_gemm_wmma<128><<<gGemm, 256, 0, stream>>>(x, pW, pb, bufB, GN, GIN, GH);
  k_scatter<<<gEdge, 256, 0, stream>>>(bufA, ei, dis, bufC);
  k_finish_stats<<<gFin, 128, 0, stream>>>(bufC, bufA, dis, b0, stats, GN);
  k_bn_relu_add<<<gNH, 256, 0, stream>>>(bufC, stats, g0, be0, bufB, bufB, GN);

  // ---- block 1: conv(x1,W1) -> bn -> relu -> + x1
  k_fill<<<gNH, 256, 0, stream>>>(bufC, 0.0f, (int)NH);
  k_fill<<<1, 256, 0, stream>>>(stats, 0.0f, 2 * GH);
  k_gemm_wmma<128><<<gGemm, 256, 0, stream>>>(bufB, W1, nullptr, bufA, GN, GH, GH);
  k_scatter<<<gEdge, 256, 0, stream>>>(bufA, ei, dis, bufC);
  k_finish_stats<<<gFin, 128, 0, stream>>>(bufC, bufA, dis, b1, stats, GN);
  k_bn_relu_add<<<gNH, 256, 0, stream>>>(bufC, stats, g1, be1, bufB, bufB, GN);

  // ---- head: out = x2 @ hW + hb   (40 cols padded to 48)
  k_gemm_wmma<48><<<gGemm, 256, 0, stream>>>(bufB, hW, hb, out, GN, GH, GOUT);
}